// MPNNNet_79637283602865
// MI455X (gfx1250) — compile-verified
//
#include <hip/hip_runtime.h>

typedef __attribute__((ext_vector_type(2))) float v2f;
typedef __attribute__((ext_vector_type(8))) float v8f;

#define HID      64
#define IN_DIM   128
#define OUT_DIM  2
#define NGRAPHS  1024

// ---------------------------------------------------------------------------
// Degree / normalization precompute
// ---------------------------------------------------------------------------
__global__ void init_deg_kernel(float* __restrict__ deg, int N) {
  int i = blockIdx.x * blockDim.x + threadIdx.x;
  if (i < N) deg[i] = 1.0f;                       // self-loop baked in
}

__global__ void deg_edges_kernel(const int* __restrict__ dst, float* __restrict__ deg, int E) {
  int e = blockIdx.x * blockDim.x + threadIdx.x;
  if (e < E) atomicAdd(&deg[dst[e]], 1.0f);
}

__global__ void dinv_kernel(const float* __restrict__ deg, float* __restrict__ dinv, int N) {
  int i = blockIdx.x * blockDim.x + threadIdx.x;
  if (i < N) dinv[i] = rsqrtf(deg[i]);
}

__global__ void enorm_kernel(const int* __restrict__ src, const int* __restrict__ dst,
                             const float* __restrict__ dinv, float* __restrict__ enorm, int E) {
  int e = blockIdx.x * blockDim.x + threadIdx.x;
  if (e < E) enorm[e] = dinv[src[e]] * dinv[dst[e]];
}

// ---------------------------------------------------------------------------
// fp32 WMMA GEMM:  C[N x 64] = A[N x K] @ W[K x 64]
// N multiple of 16 (100000 = 6250*16). One wave per 16x16 tile; block of
// 128 threads = 4 waves covering the 4 column tiles of one row tile.
// V_WMMA_F32_16X16X4_F32 fragment layout (32-bit, wave32):
//   A 16x4 : lane l: row = l&15; VGPR v holds K = 2*(l>>4) + v
//   B 4x16 : lane l: col = l&15; VGPR v holds K = 2*(l>>4) + v
//   C 16x16: lane l: col = l&15; VGPR r holds row r + 8*(l>>4)
// ---------------------------------------------------------------------------
__global__ void gemm_wmma_f32(const float* __restrict__ A,
                              const float* __restrict__ W,
                              float* __restrict__ C,
                              int K) {
  const int lane    = threadIdx.x & 31;
  const int wave    = threadIdx.x >> 5;      // 0..3 -> column tile
  const int hi      = lane >> 4;             // 0 or 1 (lane half)
  const int l15     = lane & 15;
  const int rowBase = blockIdx.x * 16;
  const int colBase = wave * 16;
  const int row     = rowBase + l15;
  const int col     = colBase + l15;

  v8f acc = {};
  for (int kk = 0; kk < K; kk += 4) {
    const int ka = kk + 2 * hi;              // even -> 8B aligned
    v2f a = *(const v2f*)(A + (size_t)row * K + ka);
    v2f b;
    b.x = W[(size_t)(ka + 0) * HID + col];
    b.y = W[(size_t)(ka + 1) * HID + col];
    acc = __builtin_amdgcn_wmma_f32_16x16x4_f32(
        /*neg_a=*/false, a, /*neg_b=*/false, b,
        /*c_mod=*/(short)0, acc, /*reuse_a=*/false, /*reuse_b=*/false);
  }

  const int rowOff = hi * 8;
#pragma unroll
  for (int r = 0; r < 8; ++r)
    C[(size_t)(rowBase + rowOff + r) * HID + colBase + l15] = acc[r];
}

// ---------------------------------------------------------------------------
// Edge scatter: agg[dst] += t[src] * enorm.  One thread per (edge, 4-feature
// chunk); 16 consecutive threads cover one edge's 64 features (coalesced
// float4 reads of t[src]).
// ---------------------------------------------------------------------------
__global__ void scatter_kernel(const float* __restrict__ t,
                               const int* __restrict__ src,
                               const int* __restrict__ dst,
                               const float* __restrict__ enorm,
                               float* __restrict__ agg, int E) {
  long long tid = (long long)blockIdx.x * blockDim.x + threadIdx.x;
  if (tid >= (long long)E * 16) return;
  const int e = (int)(tid >> 4);
  const int g = (int)(tid & 15);
  const float w = enorm[e];
  const float4 v = *(const float4*)(t + (size_t)src[e] * HID + g * 4);
  float* p = agg + (size_t)dst[e] * HID + g * 4;
  atomicAdd(p + 0, v.x * w);
  atomicAdd(p + 1, v.y * w);
  atomicAdd(p + 2, v.z * w);
  atomicAdd(p + 3, v.w * w);
}

// h = relu(agg + t/deg + bias)
__global__ void finalize_kernel(const float* __restrict__ agg, const float* __restrict__ t,
                                const float* __restrict__ deg, const float* __restrict__ bias,
                                float* __restrict__ h, int total) {
  int idx = blockIdx.x * blockDim.x + threadIdx.x;
  if (idx >= total) return;
  const int i = idx >> 6;        // node  (HID == 64)
  const int f = idx & 63;        // feature
  const float v = agg[idx] + t[idx] / deg[i] + bias[f];
  h[idx] = v > 0.0f ? v : 0.0f;
}

// ---------------------------------------------------------------------------
// Global mean pool (sum + counts via atomics), then tiny FC.
// ---------------------------------------------------------------------------
__global__ void pool_kernel(const float* __restrict__ h, const int* __restrict__ batch,
                            float* __restrict__ pooled, float* __restrict__ counts, int total) {
  int idx = blockIdx.x * blockDim.x + threadIdx.x;
  if (idx >= total) return;
  const int i = idx >> 6;
  const int f = idx & 63;
  const int g = batch[i];
  atomicAdd(&pooled[(size_t)g * HID + f], h[idx]);
  if (f == 0) atomicAdd(&counts[g], 1.0f);
}

__global__ void fc_kernel(const float* __restrict__ pooled, const float* __restrict__ counts,
                          const float* __restrict__ Wfc, const float* __restrict__ bfc,
                          float* __restrict__ out, int total) {
  int idx = blockIdx.x * blockDim.x + threadIdx.x;
  if (idx >= total) return;
  const int g = idx >> 1;
  const int o = idx & 1;
  const float inv = 1.0f / fmaxf(counts[g], 1.0f);
  float acc = bfc[o];
#pragma unroll
  for (int f = 0; f < HID; ++f)
    acc = fmaf(pooled[(size_t)g * HID + f] * inv, Wfc[f * OUT_DIM + o], acc);
  out[idx] = acc;
}

// ---------------------------------------------------------------------------
// Host-side launcher
// ---------------------------------------------------------------------------
extern "C" void kernel_launch(void* const* d_in, const int* in_sizes, int n_in,
                              void* d_out, int out_size, void* d_ws, size_t ws_size,
                              hipStream_t stream) {
  const float* x    = (const float*)d_in[0];
  const int*   edge = (const int*)  d_in[1];
  const int*   batch= (const int*)  d_in[2];
  const float* W1   = (const float*)d_in[3];
  const float* b1   = (const float*)d_in[4];
  const float* W2   = (const float*)d_in[5];
  const float* b2   = (const float*)d_in[6];
  const float* W3   = (const float*)d_in[7];
  const float* b3   = (const float*)d_in[8];
  const float* Wfc  = (const float*)d_in[9];
  const float* bfc  = (const float*)d_in[10];
  float* out = (float*)d_out;

  const int N = in_sizes[0] / IN_DIM;   // 100000 (multiple of 16)
  const int E = in_sizes[1] / 2;        // 1600000
  const int* src = edge;
  const int* dst = edge + E;

  // Workspace carve-out (256B aligned slices)
  char* ws = (char*)d_ws;
  size_t off = 0;
  auto carve = [&](size_t bytes) -> float* {
    float* p = (float*)(ws + off);
    off = (off + bytes + 255) & ~(size_t)255;
    return p;
  };
  float* deg    = carve((size_t)N * 4);
  float* dinv   = carve((size_t)N * 4);
  float* enorm  = carve((size_t)E * 4);
  float* t      = carve((size_t)N * HID * 4);
  float* agg    = carve((size_t)N * HID * 4);
  float* h      = carve((size_t)N * HID * 4);
  float* pooled = carve((size_t)NGRAPHS * HID * 4);
  float* counts = carve((size_t)NGRAPHS * 4);
  (void)ws_size; (void)n_in; (void)out_size;

  const int TB = 256;
  const int nodeBlocks = (N + TB - 1) / TB;
  const int edgeBlocks = (E + TB - 1) / TB;
  const int nhTotal    = N * HID;
  const int nhBlocks   = (nhTotal + TB - 1) / TB;
  const long long scatTotal = (long long)E * 16;
  const unsigned scatBlocks = (unsigned)((scatTotal + TB - 1) / TB);
  const int rowTiles = N / 16;

  // --- normalization precompute ---
  init_deg_kernel<<<nodeBlocks, TB, 0, stream>>>(deg, N);
  deg_edges_kernel<<<edgeBlocks, TB, 0, stream>>>(dst, deg, E);
  dinv_kernel<<<nodeBlocks, TB, 0, stream>>>(deg, dinv, N);
  enorm_kernel<<<edgeBlocks, TB, 0, stream>>>(src, dst, dinv, enorm, E);

  // --- layer 1 (K = 128) ---
  gemm_wmma_f32<<<rowTiles, 128, 0, stream>>>(x, W1, t, IN_DIM);
  hipMemsetAsync(agg, 0, (size_t)nhTotal * 4, stream);
  scatter_kernel<<<scatBlocks, TB, 0, stream>>>(t, src, dst, enorm, agg, E);
  finalize_kernel<<<nhBlocks, TB, 0, stream>>>(agg, t, deg, b1, h, nhTotal);

  // --- layer 2 (K = 64) ---
  gemm_wmma_f32<<<rowTiles, 128, 0, stream>>>(h, W2, t, HID);
  hipMemsetAsync(agg, 0, (size_t)nhTotal * 4, stream);
  scatter_kernel<<<scatBlocks, TB, 0, stream>>>(t, src, dst, enorm, agg, E);
  finalize_kernel<<<nhBlocks, TB, 0, stream>>>(agg, t, deg, b2, h, nhTotal);

  // --- layer 3 (K = 64) ---
  gemm_wmma_f32<<<rowTiles, 128, 0, stream>>>(h, W3, t, HID);
  hipMemsetAsync(agg, 0, (size_t)nhTotal * 4, stream);
  scatter_kernel<<<scatBlocks, TB, 0, stream>>>(t, src, dst, enorm, agg, E);
  finalize_kernel<<<nhBlocks, TB, 0, stream>>>(agg, t, deg, b3, h, nhTotal);

  // --- pool + FC ---
  hipMemsetAsync(pooled, 0, (size_t)NGRAPHS * HID * 4, stream);
  hipMemsetAsync(counts, 0, (size_t)NGRAPHS * 4, stream);
  pool_kernel<<<nhBlocks, TB, 0, stream>>>(h, batch, pooled, counts, nhTotal);
  fc_kernel<<<(NGRAPHS * OUT_DIM + TB - 1) / TB, TB, 0, stream>>>(pooled, counts, Wfc, bfc, out,
                                                                  NGRAPHS * OUT_DIM);
}